// Attention_50654844289068
// MI455X (gfx1250) — compile-verified
//
#include <hip/hip_runtime.h>
#include <math.h>

#define BDIM 8
#define SEQ  2048
#define DM   512
#define DK   64
#define CHUNK 512
#define NCH  (SEQ / CHUNK)

// LDS row padding (floats) to de-conflict 64 banks
#define SQ_LD (DK + 4)      // 68
#define SP_LD (CHUNK + 4)   // 516

constexpr float EPS_V = 1e-6f;

typedef __attribute__((ext_vector_type(2))) float v2f;
typedef __attribute__((ext_vector_type(8))) float v8f;

__device__ __forceinline__ v8f wmma_f32_k4(v2f a, v2f b, v8f c) {
    // D = A(16x4,f32) * B(4x16,f32) + C(16x16,f32)
    return __builtin_amdgcn_wmma_f32_16x16x4_f32(
        false, a, false, b, (short)0, c, false, false);
}

// Async global->LDS copy of one 16-byte chunk per lane (ISA: GLOBAL_LOAD_ASYNC_TO_LDS_B128)
__device__ __forceinline__ void async_ld_b128(unsigned lds_addr, const void* gptr) {
    asm volatile("global_load_async_to_lds_b128 %0, %1, off"
                 :: "v"(lds_addr), "v"((unsigned long long)gptr)
                 : "memory");
}
__device__ __forceinline__ void wait_asynccnt0() {
    asm volatile("s_wait_asynccnt 0x0" ::: "memory");
}

// ---------------------------------------------------------------------------
// Projection: Y = X(16 x 512) * W^T(512 x 64) + bias
// TRANSPOSED=false: Y laid out (B*S, 64).  TRANSPOSED=true: Y laid out (B, 64, S).
// ---------------------------------------------------------------------------
template <bool TRANSPOSED>
__global__ __launch_bounds__(128) void qkv_proj_kernel(
    const float* __restrict__ X, const float* __restrict__ W,
    const float* __restrict__ bias, float* __restrict__ Y)
{
    __shared__ float lX[16 * DM];   // 32 KB row tile of X

    const int tid  = threadIdx.x;
    const int row0 = blockIdx.x * 16;

    const float4* Xv  = (const float4*)(X + (size_t)row0 * DM);
    float4*       lXv = (float4*)lX;
    for (int i = tid; i < 16 * DM / 4; i += 128) lXv[i] = Xv[i];
    __syncthreads();

    const int wave  = tid >> 5, lane = tid & 31;
    const int n0    = wave * 16;
    const int mrow  = lane & 15;             // A-frag row
    const int khalf = (lane >> 4) << 1;      // A-frag: K=0/1 vs K=2/3
    const int kbb   = lane >> 4;             // B-frag: K=0/2 vs K=1/3
    const int nb    = n0 + (lane & 15);      // B-frag column

    v8f acc = {};
    for (int kk = 0; kk < DM; kk += 4) {
        v2f a;
        a.x = lX[mrow * DM + kk + khalf];
        a.y = lX[mrow * DM + kk + khalf + 1];
        v2f b;  // B[k][n] = W[n][k]
        b.x = W[(size_t)nb * DM + kk + kbb];
        b.y = W[(size_t)nb * DM + kk + 2 + kbb];
        acc = wmma_f32_k4(a, b, acc);
    }

    const int cm = (lane >> 4) << 3;
    const int cn = lane & 15;
    const float bn = bias[n0 + cn];
    if (TRANSPOSED) {
        const int b  = row0 / SEQ;
        const int s0 = row0 % SEQ;
        float* Yb = Y + (size_t)b * DK * SEQ;
        #pragma unroll
        for (int i = 0; i < 8; ++i)
            Yb[(size_t)(n0 + cn) * SEQ + s0 + cm + i] = acc[i] + bn;
    } else {
        #pragma unroll
        for (int i = 0; i < 8; ++i)
            Y[(size_t)(row0 + cm + i) * DK + n0 + cn] = acc[i] + bn;
    }
}

// ---------------------------------------------------------------------------
// Attention: one block per (batch, 16-query tile); flash-style over 4
// column chunks of 512.  KT is (B, 64, S) so QK^T B-fragments are coalesced.
// ---------------------------------------------------------------------------
__global__ __launch_bounds__(128) void attn_kernel(
    const float* __restrict__ Q, const float* __restrict__ KT,
    const float* __restrict__ V, const int* __restrict__ MASK,
    float* __restrict__ OUT, float* __restrict__ WGT)
{
    __shared__ float sP[16 * SP_LD];   // padded score chunk (~33 KB)
    __shared__ float sQ[16 * SQ_LD];   // padded q tile
    __shared__ float red[16 * 8];
    __shared__ float rm[16], rl[16], fac[16], chm[16 * NCH];

    const int tid  = threadIdx.x;
    const int b    = blockIdx.y;
    const int q0   = blockIdx.x * 16;
    const int wave = tid >> 5, lane = tid & 31;
    const int row  = tid & 15, seg = tid >> 4;
    const int mrow  = lane & 15;
    const int khalf = (lane >> 4) << 1;
    const int kbb   = lane >> 4;
    const int cm    = (lane >> 4) << 3;
    const int cn    = lane & 15;
    const int n0v   = wave * 16;       // PV output N tile

    if (tid < 16) { rm[tid] = -INFINITY; rl[tid] = 0.f; }

    // ---- async-stage q tile (16 x 64 f32) into padded LDS rows
    const float* qb = Q + ((size_t)b * SEQ + q0) * DK;
    #pragma unroll
    for (int it = 0; it < 2; ++it) {
        const int j   = tid + it * 128;          // float4 index, 256 total
        const int qr  = j >> 4;                  // row (64 f32 = 16 float4 per row)
        const int qc  = (j & 15) << 2;           // col in floats
        const unsigned lds = (unsigned)(uintptr_t)&sQ[qr * SQ_LD + qc];
        async_ld_b128(lds, qb + (size_t)qr * DK + qc);
    }
    wait_asynccnt0();
    __syncthreads();

    const float* ktb  = KT + (size_t)b * DK * SEQ;
    const float* vb   = V  + (size_t)b * SEQ * DK;
    const int*   mb   = MASK + (size_t)b * SEQ * SEQ + (size_t)q0 * SEQ;
    float*       wout = WGT  + (size_t)b * SEQ * SEQ + (size_t)q0 * SEQ;
    const float scale = 0.125f;   // 64^-0.5

    v8f acc = {};   // running P*V accumulator (16x16 tile per wave)

    for (int ch = 0; ch < NCH; ++ch) {
        const int col0 = ch * CHUNK;

        // prefetch next chunk's mask rows into L2 while we compute this one
        if (ch + 1 < NCH)
            __builtin_prefetch(mb + (size_t)row * SEQ + (ch + 1) * CHUNK
                                  + seg * (CHUNK / 8), 0, 0);

        // ---- scores: S_tile = q_tile * K^T chunk (B-frags coalesced from KT)
        for (int nt = wave; nt < CHUNK / 16; nt += 4) {
            const int n0 = nt * 16;
            const float* kcol = ktb + col0 + n0 + cn;   // lane-consecutive
            v8f s = {};
            #pragma unroll
            for (int kk = 0; kk < DK; kk += 4) {
                v2f a;
                a.x = sQ[mrow * SQ_LD + kk + khalf];
                a.y = sQ[mrow * SQ_LD + kk + khalf + 1];
                v2f bf;   // B[k][n] = KT[k][col]
                bf.x = kcol[(size_t)(kk + kbb) * SEQ];
                bf.y = kcol[(size_t)(kk + 2 + kbb) * SEQ];
                s = wmma_f32_k4(a, bf, s);
            }
            #pragma unroll
            for (int i = 0; i < 8; ++i) {
                const int m = cm + i, n = n0 + cn;
                float sc = s[i] * scale;
                sc = (sc == 0.f) ? EPS_V : sc;
                if (mb[(size_t)m * SEQ + col0 + n] == 0) sc = EPS_V;
                sP[m * SP_LD + n] = sc;
            }
        }
        __syncthreads();

        // ---- chunk row max (8 threads per row)
        float cmx = -INFINITY;
        for (int c = seg; c < CHUNK; c += 8) cmx = fmaxf(cmx, sP[row * SP_LD + c]);
        red[row * 8 + seg] = cmx;
        __syncthreads();
        if (tid < 16) {
            float mc = red[tid * 8];
            #pragma unroll
            for (int j = 1; j < 8; ++j) mc = fmaxf(mc, red[tid * 8 + j]);
            const float mold = rm[tid];
            const float mnew = fmaxf(mold, mc);
            fac[tid] = __expf(mold - mnew);      // 0 on first chunk (exp(-inf))
            rm[tid]  = mnew;
            chm[tid * NCH + ch] = mnew;
        }
        __syncthreads();

        // ---- exponentiate in-place + partial row sums
        const float mnew = rm[row];
        float psum = 0.f;
        for (int c = seg; c < CHUNK; c += 8) {
            const float e = __expf(sP[row * SP_LD + c] - mnew);
            sP[row * SP_LD + c] = e;
            psum += e;
        }
        red[row * 8 + seg] = psum;
        __syncthreads();
        if (tid < 16) {
            float s = 0.f;
            #pragma unroll
            for (int j = 0; j < 8; ++j) s += red[tid * 8 + j];
            rl[tid] = rl[tid] * fac[tid] + s;
        }

        // ---- stream unnormalized (shifted-exp) weights to global once
        for (int i = tid; i < 16 * CHUNK; i += 128)
            wout[(size_t)(i >> 9) * SEQ + col0 + (i & (CHUNK - 1))]
                = sP[(i >> 9) * SP_LD + (i & (CHUNK - 1))];

        // ---- rescale accumulator, accumulate P_chunk * V_chunk
        #pragma unroll
        for (int i = 0; i < 8; ++i) acc[i] *= fac[cm + i];
        for (int kk = 0; kk < CHUNK; kk += 4) {
            v2f a;
            a.x = sP[mrow * SP_LD + kk + khalf];
            a.y = sP[mrow * SP_LD + kk + khalf + 1];
            v2f bf;  // B[k][n] = v[col0+k][n], lane-consecutive in n
            const float* vr = vb + (size_t)(col0 + kk) * DK + n0v + cn;
            bf.x = vr[(size_t)kbb * DK];
            bf.y = vr[(size_t)(2 + kbb) * DK];
            acc = wmma_f32_k4(a, bf, acc);
        }
        __syncthreads();   // sP reused next chunk
    }

    // ---- final attention output: acc is at scale rm (final max); divide by l
    float* aout = OUT + ((size_t)b * SEQ + q0) * DK;
    #pragma unroll
    for (int i = 0; i < 8; ++i)
        aout[(size_t)(cm + i) * DK + n0v + cn] = acc[i] / rl[cm + i];

    // ---- weight normalization: identical thread<->address mapping as the
    // chunk writes above, so each thread RMWs only its own stores.
    for (int ch = 0; ch < NCH; ++ch) {
        const int col0 = ch * CHUNK;
        for (int i = tid; i < 16 * CHUNK; i += 128) {
            const int m = i >> 9;
            const float f = __expf(chm[m * NCH + ch] - rm[m]) / rl[m];
            wout[(size_t)m * SEQ + col0 + (i & (CHUNK - 1))] *= f;
        }
    }
}

// ---------------------------------------------------------------------------
extern "C" void kernel_launch(void* const* d_in, const int* in_sizes, int n_in,
                              void* d_out, int out_size, void* d_ws, size_t ws_size,
                              hipStream_t stream) {
    const float* query = (const float*)d_in[0];
    const float* key   = (const float*)d_in[1];
    const float* value = (const float*)d_in[2];
    const int*   mask  = (const int*)d_in[3];
    const float* Wq = (const float*)d_in[4];
    const float* bq = (const float*)d_in[5];
    const float* Wk = (const float*)d_in[6];
    const float* bk = (const float*)d_in[7];
    const float* Wv = (const float*)d_in[8];
    const float* bv = (const float*)d_in[9];

    float* qp  = (float*)d_ws;                      // (B,S,64)  4 MB
    float* ktp = qp  + (size_t)BDIM * SEQ * DK;     // (B,64,S)  4 MB (transposed)
    float* vp  = ktp + (size_t)BDIM * SEQ * DK;     // (B,S,64)  4 MB

    float* out_attn = (float*)d_out;                          // (B,S,64)
    float* out_wgt  = out_attn + (size_t)BDIM * SEQ * DK;     // (B,S,S)

    dim3 pgrid(BDIM * SEQ / 16), pblk(128);
    qkv_proj_kernel<false><<<pgrid, pblk, 0, stream>>>(query, Wq, bq, qp);
    qkv_proj_kernel<true ><<<pgrid, pblk, 0, stream>>>(key,   Wk, bk, ktp);
    qkv_proj_kernel<false><<<pgrid, pblk, 0, stream>>>(value, Wv, bv, vp);

    dim3 agrid(SEQ / 16, BDIM), ablk(128);
    attn_kernel<<<agrid, ablk, 0, stream>>>(qp, ktp, vp, mask, out_attn, out_wgt);
}